// VQVAE_55989193671350
// MI455X (gfx1250) — compile-verified
//
#include <hip/hip_runtime.h>
#include <stdint.h>

// VQ-VAE codebook quantization for MI455X (gfx1250), wave32.
// - exact-fp32 V_WMMA_F32_16X16X4_F32 for the x . e term (argmin fidelity)
// - codebook staged through LDS by the Tensor Data Mover (TENSOR_LOAD_TO_LDS),
//   double-buffered, with TDM row padding chosen for conflict-free ds_load_b64.

typedef __attribute__((ext_vector_type(2))) float v2f;
typedef __attribute__((ext_vector_type(8))) float v8f;
typedef __attribute__((ext_vector_type(4))) unsigned int v4u;
typedef __attribute__((ext_vector_type(4))) int v4i;
typedef __attribute__((ext_vector_type(8))) int v8i;

#define C_DIM          64
#define CHUNK_ROWS     64          // codebook rows per TDM chunk (4 k-tiles)
#define ROW_STRIDE_DW  68          // 64 data DWORDs + 4 pad DWORDs (TDM pad engine)
#define ROWS_PER_BLOCK 32          // two 16-row M-tiles per block
#define WAVES          8

// ---- kernel 1: e_sq[n] = sum_c codebook[n][c]^2 -------------------------
__global__ __launch_bounds__(256) void vq_esq_kernel(
    const float* __restrict__ cb, float* __restrict__ esq, int K) {
  int row = blockIdx.x * blockDim.x + threadIdx.x;
  if (row >= K) return;
  const float4* p = (const float4*)(cb + (size_t)row * C_DIM);
  float s = 0.f;
#pragma unroll
  for (int i = 0; i < C_DIM / 4; ++i) {
    float4 v = p[i];
    s += v.x * v.x + v.y * v.y + v.z * v.z + v.w * v.w;
  }
  esq[row] = s;
}

// ---- TDM: DMA one 64x64 f32 codebook tile into LDS with row padding -----
// D# per CDNA5 ISA 8.3-8.6: group0 = {ctl, lds_addr, global_addr, type},
// group1 = dims/strides/tile + pad config. Groups 2/3 zero (2D tensor).
// This toolchain exposes the 6-arg builtin (extra int32x8 group before cpol).
__device__ __forceinline__ void tdm_load_chunk(const float* cb, int K,
                                               int chunk, uint32_t lds_off) {
  uint64_t ga = (uint64_t)(uintptr_t)(cb + (size_t)chunk * CHUNK_ROWS * C_DIM);
  v4u g0;
  g0[0] = 1u;                                       // count=1, user mode
  g0[1] = lds_off;                                  // lds_addr (bytes)
  g0[2] = (uint32_t)ga;                             // global_addr[31:0]
  g0[3] = (uint32_t)((ga >> 32) & 0x1FFFFFFu)       // global_addr[56:32]
        | (2u << 30);                               // type = 2 ("image")
  v8i g1;
  g1[0] = (2 << 16)                                 // data_size = 4 bytes
        | (1 << 20)                                 // pad_enable
        | (5 << 22)                                 // pad_interval: 2^(5+1)=64 DW (one row)
        | (3 << 25);                                // pad_amount: 3+1 = 4 DWORDs
  g1[1] = (C_DIM & 0xFFFF) << 16;                   // tensor_dim0[15:0]
  g1[2] = (C_DIM >> 16) | ((K & 0xFFFF) << 16);     // dim0 hi | tensor_dim1 lo
  g1[3] = ((unsigned)K >> 16) | (C_DIM << 16);      // dim1 hi | tile_dim0 = 64
  g1[4] = CHUNK_ROWS;                               // tile_dim1 = 64 (tile_dim2 = 0)
  g1[5] = C_DIM;                                    // tensor_dim0_stride lo32
  g1[6] = 0;
  g1[7] = 0;
  v4i z4 = {0, 0, 0, 0};
  v8i z8 = {0, 0, 0, 0, 0, 0, 0, 0};
  __builtin_amdgcn_tensor_load_to_lds(g0, g1, z4, z4, z8, 0);
}

// ---- kernel 2: 32-row tile: TDM-staged WMMA distances, argmin, gather ---
__global__ __launch_bounds__(256) void vq_argmin_gather_kernel(
    const float* __restrict__ x,      // [N][64]
    const float* __restrict__ cb,     // [K][64]
    const float* __restrict__ esq,    // [K]
    float* __restrict__ out,          // [N][64]
    int* __restrict__ idx_out,        // [N]
    int K) {
  __shared__ float cbuf[2][CHUNK_ROWS * ROW_STRIDE_DW];  // 2 x 17KB staging
  __shared__ float s_val[WAVES][16];
  __shared__ int   s_idx[WAVES][16];
  __shared__ int   s_final[ROWS_PER_BLOCK];

  const int tid    = threadIdx.x;
  const int wave   = tid >> 5;
  const int lane   = tid & 31;
  const int laneLo = lane & 15;
  const int hi     = lane >> 4;        // 0: K=0,1 half ; 1: K=2,3 half
  const int mt     = wave >> 2;        // which 16-row M-tile (0/1)
  const int wq     = wave & 3;         // which k-tile quarter of each chunk
  const int mbase  = blockIdx.x * ROWS_PER_BLOCK + mt * 16;

  // Preload A fragments (16x64 query tile, invariant over the column loop).
  v2f a[16];
  {
    const float* xrow = x + (size_t)(mbase + laneLo) * C_DIM + 2 * hi;
#pragma unroll
    for (int s = 0; s < 16; ++s)
      a[s] = *(const v2f*)(xrow + 4 * s);
  }

  float minv[8];
  int   mini[8];
#pragma unroll
  for (int r = 0; r < 8; ++r) { minv[r] = 3.4e38f; mini[r] = 0x7fffffff; }

  const uint32_t lds0 = (uint32_t)(uintptr_t)(void*)&cbuf[0][0];
  const uint32_t lds1 = (uint32_t)(uintptr_t)(void*)&cbuf[1][0];
  const int nchunks = K / CHUNK_ROWS;

  if (wave == 0) tdm_load_chunk(cb, K, 0, lds0);   // prime buffer 0

  for (int c = 0; c < nchunks; ++c) {
    if (wave == 0) {
      if (c + 1 < nchunks) {
        tdm_load_chunk(cb, K, c + 1, ((c + 1) & 1) ? lds1 : lds0);
        __builtin_amdgcn_s_wait_tensorcnt(1);      // chunk c has landed
      } else {
        __builtin_amdgcn_s_wait_tensorcnt(0);
      }
    }
    __syncthreads();                               // release chunk c to all waves

    // B fragments from LDS: row stride 68 DW -> lo lanes hit banks {4L+c,4L+c+1},
    // hi lanes {4L+c+2,4L+c+3}: all 64 banks exactly once, 8B aligned.
    const int   nloc = wq * 16 + laneLo;           // row within chunk
    const float* bp  = &cbuf[c & 1][nloc * ROW_STRIDE_DW + 2 * hi];
    v2f b[16];
#pragma unroll
    for (int s = 0; s < 16; ++s)
      b[s] = *(const v2f*)(bp + 4 * s);

    // Two independent accumulator chains for matrix-pipe ILP.
    v8f acc0 = {}, acc1 = {};
#pragma unroll
    for (int s = 0; s < 16; s += 2) {
      acc0 = __builtin_amdgcn_wmma_f32_16x16x4_f32(false, a[s],     false, b[s],
                                                   (short)0, acc0, false, false);
      acc1 = __builtin_amdgcn_wmma_f32_16x16x4_f32(false, a[s + 1], false, b[s + 1],
                                                   (short)0, acc1, false, false);
    }

    const int   n = c * CHUNK_ROWS + nloc;
    const float e = esq[n];
#pragma unroll
    for (int r = 0; r < 8; ++r) {
      float dot = acc0[r] + acc1[r];
      float d   = __builtin_fmaf(-2.0f, dot, e);
      if (d < minv[r]) { minv[r] = d; mini[r] = n; }  // n increases per wave
    }
    __syncthreads();   // all waves done reading cbuf[c&1] before it is refilled
  }

  // Reduce (val,idx) across the 16-lane N-groups.
#pragma unroll
  for (int r = 0; r < 8; ++r) {
    float v = minv[r];
    int   i = mini[r];
#pragma unroll
    for (int m = 8; m >= 1; m >>= 1) {
      float ov = __shfl_xor(v, m);
      int   oi = __shfl_xor(i, m);
      if (ov < v || (ov == v && oi < i)) { v = ov; i = oi; }
    }
    if (laneLo == 0) {                 // lane 0 -> row r ; lane 16 -> row r+8
      s_val[wave][r + 8 * hi] = v;
      s_idx[wave][r + 8 * hi] = i;
    }
  }
  __syncthreads();

  // Cross-wave reduction: 4 waves per M-tile; one thread per query row.
  if (tid < ROWS_PER_BLOCK) {
    const int mtF = tid >> 4;          // which M-tile
    const int w0  = mtF * 4;
    float v = s_val[w0][tid & 15];
    int   i = s_idx[w0][tid & 15];
#pragma unroll
    for (int w = 1; w < 4; ++w) {
      float ov = s_val[w0 + w][tid & 15];
      int   oi = s_idx[w0 + w][tid & 15];
      if (ov < v || (ov == v && oi < i)) { v = ov; i = oi; }
    }
    s_final[tid] = i;
    idx_out[blockIdx.x * ROWS_PER_BLOCK + tid] = i;
  }
  __syncthreads();

  // Gather winners: out = codebook[idx] (straight-through forward == q).
  // 8 threads per row, 2x float4 each -> 256B/row coalesced.
  {
    const int rlocal = tid >> 3;
    const int e8     = (tid & 7) * 8;
    const int code   = s_final[rlocal];
    const float* src = cb + (size_t)code * C_DIM + e8;
    float*       dst = out + (size_t)(blockIdx.x * ROWS_PER_BLOCK + rlocal) * C_DIM + e8;
    *(float4*)(dst)     = *(const float4*)(src);
    *(float4*)(dst + 4) = *(const float4*)(src + 4);
  }
}

extern "C" void kernel_launch(void* const* d_in, const int* in_sizes, int n_in,
                              void* d_out, int out_size, void* d_ws, size_t ws_size,
                              hipStream_t stream) {
  const float* x  = (const float*)d_in[0];   // encoded [B,W,H,C] -> [N][64]
  const float* cb = (const float*)d_in[1];   // codebook [K][64]
  const int N = in_sizes[0] / C_DIM;         // 65536
  const int K = in_sizes[1] / C_DIM;         // 4096

  float* out    = (float*)d_out;                       // [N][64] floats
  int*   idxout = (int*)(out + (size_t)N * C_DIM);     // [N] int32, concatenated
  float* esq    = (float*)d_ws;                        // [K] floats scratch

  vq_esq_kernel<<<(K + 255) / 256, 256, 0, stream>>>(cb, esq, K);
  vq_argmin_gather_kernel<<<N / ROWS_PER_BLOCK, 256, 0, stream>>>(
      x, cb, esq, out, idxout, K);
}